// HeteroSAGE_32770600468607
// MI455X (gfx1250) — compile-verified
//
#include <hip/hip_runtime.h>

#define DD 128

typedef float v2f __attribute__((ext_vector_type(2)));
typedef float v8f __attribute__((ext_vector_type(8)));

// ---------------------------------------------------------------------------
// Count in-degree per destination node (float, reused all 3 layers).
// ---------------------------------------------------------------------------
__global__ void hs_count(const int* __restrict__ dst, float* __restrict__ cnt, int E) {
    int i = blockIdx.x * blockDim.x + threadIdx.x;
    if (i < E) atomicAdd(&cnt[dst[i]], 1.0f);
}

// ---------------------------------------------------------------------------
// Repack one 128x128 row-major W into WMMA-fragment-contiguous layout:
// for frag f=kb/4, tile n, lane (lo=lane&15, hi=lane>>4):
//   pair = ( W[kb+hi][n*16+lo], W[kb+hi+2][n*16+lo] )
// stored at float2 index  n*1024 + f*32 + lane  -> B frag = one coalesced b64.
// ---------------------------------------------------------------------------
__global__ void hs_repack(const float* __restrict__ W, float* __restrict__ Wp) {
    int t = blockIdx.x * blockDim.x + threadIdx.x;   // 0..8191
    if (t >= 8192) return;
    int lane = t & 31;
    int f    = (t >> 5) & 31;
    int n    = t >> 10;
    int lo = lane & 15, hi = lane >> 4;
    int kb = f * 4;
    float2 v;
    v.x = W[(kb + hi) * DD + n * 16 + lo];
    v.y = W[(kb + hi + 2) * DD + n * 16 + lo];
    ((float2*)Wp)[(size_t)n * 1024 + f * 32 + lane] = v;
}

// ---------------------------------------------------------------------------
// Fused edge kernel: one wave (32 lanes) per edge; lane owns 4 contiguous
// columns (float4 loads of h/bem/Wem rows; 4.6KB Wem + L2-resident h).
//   val[c] = h_src[src][c] + (ea[e] @ Wem + bem)[c]; atomicAdd into s[dst][c]
// ---------------------------------------------------------------------------
__global__ void hs_scatter(const float* __restrict__ hsrc,
                           const float* __restrict__ ea,
                           const float* __restrict__ Wem,
                           const float* __restrict__ bem,
                           const int*   __restrict__ src,
                           const int*   __restrict__ dst,
                           float* __restrict__ s, int E) {
    int t = blockIdx.x * blockDim.x + threadIdx.x;
    int e = t >> 5;
    int lane = t & 31;
    if (e >= E) return;
    int sn = src[e], dn = dst[e];
    float a[9];
#pragma unroll
    for (int j = 0; j < 9; ++j) a[j] = ea[(size_t)e * 9 + j];   // 36B stride: scalar
    int c = lane * 4;
    float4 v  = *(const float4*)(bem + c);
#pragma unroll
    for (int j = 0; j < 9; ++j) {
        float4 w = *(const float4*)(Wem + j * DD + c);
        v.x = fmaf(a[j], w.x, v.x); v.y = fmaf(a[j], w.y, v.y);
        v.z = fmaf(a[j], w.z, v.z); v.w = fmaf(a[j], w.w, v.w);
    }
    float4 hv = *(const float4*)(hsrc + (size_t)sn * DD + c);
    v.x += hv.x; v.y += hv.y; v.z += hv.z; v.w += hv.w;
    float* sr = s + (size_t)dn * DD + c;
    atomicAdd(sr + 0, v.x);
    atomicAdd(sr + 1, v.y);
    atomicAdd(sr + 2, v.z);
    atomicAdd(sr + 3, v.w);
}

// ---------------------------------------------------------------------------
// s[node][c] /= max(count,1)  (float4 / b128 path)
// ---------------------------------------------------------------------------
__global__ void hs_meandiv(float* __restrict__ s, const float* __restrict__ cnt, int total4) {
    int i = blockIdx.x * blockDim.x + threadIdx.x;
    if (i < total4) {
        float inv = 1.0f / fmaxf(cnt[i >> 5], 1.0f);
        float4 v = ((const float4*)s)[i];
        v.x *= inv; v.y *= inv; v.z *= inv; v.w *= inv;
        ((float4*)s)[i] = v;
    }
}

// ---------------------------------------------------------------------------
// Fused dual-GEMM + bias + LayerNorm with fp32 WMMA (V_WMMA_F32_16X16X4_F32).
// One wave computes a 32x128 stripe (two 16-row M-blocks x 8 N-tiles = 16
// v8f accumulators). Per K-step: ALL 20 b64 fragment loads are issued first
// (one clause, partial waitcnts), then 32 WMMAs, interleaved so consecutive
// WMMAs never share an accumulator.
//   out = mean @ Wl + bl + h @ Wr ; LayerNorm ; in-place update of h.
// EXEC must be all-ones for WMMA: waves exit whole; per-lane guards only
// after the last WMMA.
// ---------------------------------------------------------------------------
__global__ __launch_bounds__(256)
void hs_sage_gemm_ln(float* __restrict__ h,           // [N,128] in-out (x_dst)
                     const float* __restrict__ mean,  // [N,128] aggregated mean
                     const float* __restrict__ Wlp,   // packed Wl (16384 floats)
                     const float* __restrict__ bl,    // [128]
                     const float* __restrict__ Wrp,   // packed Wr
                     const float* __restrict__ gam,
                     const float* __restrict__ bet,
                     int N) {
    int wave = blockIdx.x * 8 + (threadIdx.x >> 5);
    int row0 = wave * 32;
    if (row0 >= N) return;                 // wave-uniform exit (EXEC stays full)
    int lane = threadIdx.x & 31;
    int lo = lane & 15, hi = lane >> 4;

    v8f acc[2][8];
#pragma unroll
    for (int n = 0; n < 8; ++n) {          // init C with broadcast bias bl[N]
        float b = bl[n * 16 + lo];
        acc[0][n] = (v8f){b, b, b, b, b, b, b, b};
        acc[1][n] = (v8f){b, b, b, b, b, b, b, b};
    }

    int ar0 = row0 + lo;      if (ar0 > N - 1) ar0 = N - 1;
    int ar1 = row0 + 16 + lo; if (ar1 > N - 1) ar1 = N - 1;
    const float* a1p0 = mean + (size_t)ar0 * DD + 2 * hi;
    const float* a2p0 = h    + (size_t)ar0 * DD + 2 * hi;
    const float* a1p1 = mean + (size_t)ar1 * DD + 2 * hi;
    const float* a2p1 = h    + (size_t)ar1 * DD + 2 * hi;
    const v2f* blp = (const v2f*)Wlp + lane;   // + f*32 + n*1024
    const v2f* brp = (const v2f*)Wrp + lane;

    for (int kb = 0; kb < DD; kb += 4) {
        int f32off = (kb >> 2) * 32;
        // ---- load phase: 4 A frags + 16 B frags, all issued before any use
        v2f a1m0 = *(const v2f*)(a1p0 + kb);
        v2f a2m0 = *(const v2f*)(a2p0 + kb);
        v2f a1m1 = *(const v2f*)(a1p1 + kb);
        v2f a2m1 = *(const v2f*)(a2p1 + kb);
        v2f b1[8], b2[8];
#pragma unroll
        for (int n = 0; n < 8; ++n) b1[n] = blp[f32off + n * 1024];
#pragma unroll
        for (int n = 0; n < 8; ++n) b2[n] = brp[f32off + n * 1024];
        // ---- compute phase: 32 WMMAs, M-block interleaved
#pragma unroll
        for (int n = 0; n < 8; ++n) {
            acc[0][n] = __builtin_amdgcn_wmma_f32_16x16x4_f32(
                false, a1m0, false, b1[n], (short)0, acc[0][n], false, false);
            acc[1][n] = __builtin_amdgcn_wmma_f32_16x16x4_f32(
                false, a1m1, false, b1[n], (short)0, acc[1][n], false, false);
            acc[0][n] = __builtin_amdgcn_wmma_f32_16x16x4_f32(
                false, a2m0, false, b2[n], (short)0, acc[0][n], false, false);
            acc[1][n] = __builtin_amdgcn_wmma_f32_16x16x4_f32(
                false, a2m1, false, b2[n], (short)0, acc[1][n], false, false);
        }
    }

    float gv[8], bv[8];
#pragma unroll
    for (int n = 0; n < 8; ++n) { gv[n] = gam[n * 16 + lo]; bv[n] = bet[n * 16 + lo]; }

    // LayerNorm in registers: C-tile row m spans 16 lanes of a half-wave x 8
    // n-tiles -> reduce 8 regs locally + shfl_xor (masks 1..8 stay in-half).
#pragma unroll
    for (int mb = 0; mb < 2; ++mb) {
#pragma unroll
        for (int r = 0; r < 8; ++r) {
            float sm = 0.f, sq = 0.f;
#pragma unroll
            for (int n = 0; n < 8; ++n) {
                float v = acc[mb][n][r];
                sm += v; sq = fmaf(v, v, sq);
            }
#pragma unroll
            for (int m = 1; m < 16; m <<= 1) {
                sm += __shfl_xor(sm, m, 32);
                sq += __shfl_xor(sq, m, 32);
            }
            float mu  = sm * (1.0f / 128.0f);
            float var = sq * (1.0f / 128.0f) - mu * mu;
            float is  = rsqrtf(var + 1e-5f);
            int gr = row0 + mb * 16 + r + 8 * hi;
            if (gr < N) {
                float* out = h + (size_t)gr * DD + lo;
#pragma unroll
                for (int n = 0; n < 8; ++n)
                    out[n * 16] = (acc[mb][n][r] - mu) * is * gv[n] + bv[n];
            }
        }
    }
}

// ---------------------------------------------------------------------------
// Host orchestration (graph-capture safe: only async ops on `stream`).
// Workspace: h_a, h_b (updated in-place), s_a, s_b, cnt_a, cnt_b,
// 12 packed weight matrices (768KB)  -> ~206 MB total.
// ---------------------------------------------------------------------------
extern "C" void kernel_launch(void* const* d_in, const int* in_sizes, int n_in,
                              void* d_out, int out_size, void* d_ws, size_t ws_size,
                              hipStream_t stream) {
    (void)n_in; (void)out_size; (void)ws_size;
    const float* x_a    = (const float*)d_in[0];
    const float* x_b    = (const float*)d_in[1];
    const float* ea_ab  = (const float*)d_in[2];
    const float* ea_ba  = (const float*)d_in[3];
    const float* Wem_ab = (const float*)d_in[4];
    const float* bem_ab = (const float*)d_in[5];
    const float* Wem_ba = (const float*)d_in[6];
    const float* bem_ba = (const float*)d_in[7];
    const float* Wl_ab  = (const float*)d_in[8];
    const float* bl_ab  = (const float*)d_in[9];
    const float* Wr_ab  = (const float*)d_in[10];
    const float* Wl_ba  = (const float*)d_in[11];
    const float* bl_ba  = (const float*)d_in[12];
    const float* Wr_ba  = (const float*)d_in[13];
    const float* g_a    = (const float*)d_in[14];
    const float* be_a   = (const float*)d_in[15];
    const float* g_b    = (const float*)d_in[16];
    const float* be_b   = (const float*)d_in[17];
    const int*   src_ab = (const int*)d_in[18];
    const int*   dst_ab = (const int*)d_in[19];
    const int*   src_ba = (const int*)d_in[20];
    const int*   dst_ba = (const int*)d_in[21];

    int Na = in_sizes[0] / DD;
    int Nb = in_sizes[1] / DD;
    int E  = in_sizes[18];

    size_t NaD = (size_t)Na * DD, NbD = (size_t)Nb * DD;
    const size_t WP = 16384;                 // floats per packed matrix
    float* h_a   = (float*)d_ws;
    float* h_b   = h_a + NaD;
    float* s_a   = h_b + NbD;
    float* s_b   = s_a + NaD;
    float* cnt_a = s_b + NbD;
    float* cnt_b = cnt_a + Na;
    float* wp    = cnt_b + Nb;               // 12 * 16384 floats
    float* Wlp_ab = wp;                      // [3][16384]
    float* Wrp_ab = wp + 3 * WP;
    float* Wlp_ba = wp + 6 * WP;
    float* Wrp_ba = wp + 9 * WP;

    hipMemcpyAsync(h_a, x_a, NaD * sizeof(float), hipMemcpyDeviceToDevice, stream);
    hipMemcpyAsync(h_b, x_b, NbD * sizeof(float), hipMemcpyDeviceToDevice, stream);
    hipMemsetAsync(cnt_a, 0, (size_t)(Na + Nb) * sizeof(float), stream);

    // One-time weight repack (12 x 64KB) for coalesced b64 B-fragments.
    for (int l = 0; l < 3; ++l) {
        hs_repack<<<32, 256, 0, stream>>>(Wl_ab + (size_t)l * WP, Wlp_ab + (size_t)l * WP);
        hs_repack<<<32, 256, 0, stream>>>(Wr_ab + (size_t)l * WP, Wrp_ab + (size_t)l * WP);
        hs_repack<<<32, 256, 0, stream>>>(Wl_ba + (size_t)l * WP, Wlp_ba + (size_t)l * WP);
        hs_repack<<<32, 256, 0, stream>>>(Wr_ba + (size_t)l * WP, Wrp_ba + (size_t)l * WP);
    }

    int cb = (E + 255) / 256;
    hs_count<<<cb, 256, 0, stream>>>(dst_ab, cnt_b, E);
    hs_count<<<cb, 256, 0, stream>>>(dst_ba, cnt_a, E);

    int sb = (int)(((size_t)E * 32 + 255) / 256);
    int ga = ((Na + 31) / 32 + 7) / 8;
    int gb = ((Nb + 31) / 32 + 7) / 8;

    for (int l = 0; l < 3; ++l) {
        hipMemsetAsync(s_a, 0, (NaD + NbD) * sizeof(float), stream);   // s_a|s_b contiguous
        // Both scatters read OLD h_a/h_b; both GEMMs after (stream-ordered),
        // so in-place h updates are safe.
        hs_scatter<<<sb, 256, 0, stream>>>(h_a, ea_ab, Wem_ab, bem_ab, src_ab, dst_ab, s_b, E);
        hs_scatter<<<sb, 256, 0, stream>>>(h_b, ea_ba, Wem_ba, bem_ba, src_ba, dst_ba, s_a, E);
        hs_meandiv<<<(int)((NbD / 4 + 255) / 256), 256, 0, stream>>>(s_b, cnt_b, (int)(NbD / 4));
        hs_meandiv<<<(int)((NaD / 4 + 255) / 256), 256, 0, stream>>>(s_a, cnt_a, (int)(NaD / 4));
        hs_sage_gemm_ln<<<gb, 256, 0, stream>>>(h_b, s_b,
            Wlp_ab + (size_t)l * WP, bl_ab + (size_t)l * DD,
            Wrp_ab + (size_t)l * WP, g_b, be_b, Nb);
        hs_sage_gemm_ln<<<ga, 256, 0, stream>>>(h_a, s_a,
            Wlp_ba + (size_t)l * WP, bl_ba + (size_t)l * DD,
            Wrp_ba + (size_t)l * WP, g_a, be_a, Na);
    }

    // Output: stack([h_a, h_b]) -> [2, N, 128] flat
    hipMemcpyAsync(d_out, h_a, NaD * sizeof(float), hipMemcpyDeviceToDevice, stream);
    hipMemcpyAsync((float*)d_out + NaD, h_b, NbD * sizeof(float), hipMemcpyDeviceToDevice, stream);
}